// GINEConvMLP_38173669327254
// MI455X (gfx1250) — compile-verified
//
#include <hip/hip_runtime.h>
#include <hip/hip_bf16.h>

typedef float v2f __attribute__((ext_vector_type(2)));
typedef float v8f __attribute__((ext_vector_type(8)));

#define N_NODES 50000
#define N_EDGES 800000
#define DIM 64
#define WS 68   // LDS row stride (floats): 64 + 4 pad -> conflict-free b64 LDS reads

// ---------------- Kernel 1: zero the aggregation buffer ----------------
__global__ void gine_zero_kernel(float* __restrict__ aggr, int n4) {
    int i = blockIdx.x * blockDim.x + threadIdx.x;
    if (i < n4) {
        ((float4*)aggr)[i] = make_float4(0.f, 0.f, 0.f, 0.f);
    }
}

// ---------------- Kernel 2: edge message + scatter-add ----------------
// 16 threads per edge, float4 per thread. msg = relu(x[src] + edge_attr),
// atomic f32 add into aggr[dst].
__global__ void gine_edge_kernel(const float* __restrict__ x,
                                 const long long* __restrict__ ei,
                                 const float* __restrict__ ea,
                                 float* __restrict__ aggr) {
    long long t = (long long)blockIdx.x * blockDim.x + threadIdx.x;
    int e = (int)(t >> 4);
    if (e >= N_EDGES) return;
    int q = ((int)t & 15) << 2;                 // float offset 0..60

    long long s = ei[e];
    long long d = ei[N_EDGES + e];

    float4 xv = *(const float4*)(x + s * DIM + q);
    float4 ev = *(const float4*)(ea + (long long)e * DIM + q);

    float m0 = fmaxf(xv.x + ev.x, 0.f);
    float m1 = fmaxf(xv.y + ev.y, 0.f);
    float m2 = fmaxf(xv.z + ev.z, 0.f);
    float m3 = fmaxf(xv.w + ev.w, 0.f);

    float* ap = aggr + d * DIM + q;
    unsafeAtomicAdd(ap + 0, m0);
    unsafeAtomicAdd(ap + 1, m1);
    unsafeAtomicAdd(ap + 2, m2);
    unsafeAtomicAdd(ap + 3, m3);
}

// ---------------- Kernel 3: fused h = x + aggr, 3-layer MLP via WMMA ----------------
// Block: 256 threads = 8 waves. Block handles 128 rows; each wave owns a 16x64 tile.
// Per layer: stage W (64x64, row-major [out][in]) + bias into LDS, then
// out[m][n] = sum_k h[m][k] * W[n][k] + b[n]   via V_WMMA_F32_16X16X4_F32.
__global__ void gine_mlp_kernel(const float* __restrict__ x,
                                const float* __restrict__ aggr,
                                const float* __restrict__ W1, const float* __restrict__ b1,
                                const float* __restrict__ W2, const float* __restrict__ b2,
                                const float* __restrict__ W3, const float* __restrict__ b3,
                                float* __restrict__ out) {
    __shared__ float sW[DIM * WS];        // 17408 B: current layer's weights (padded)
    __shared__ float sB[DIM];             //   256 B: current layer's bias
    __shared__ float sH[8][16 * WS];      // 34816 B: per-wave 16x64 h tile (padded)

    const int tid   = threadIdx.x;
    const int wid   = tid >> 5;
    const int lane  = tid & 31;
    const int l15   = lane & 15;
    const int lhalf = lane >> 4;          // 0: lanes 0-15, 1: lanes 16-31
    const int base  = blockIdx.x * 128;

    // ---- load h0 = x + aggr into per-wave LDS tiles (float4, coalesced) ----
    for (int idx = tid; idx < 128 * 16; idx += 256) {
        int r  = idx >> 4;                // row in block 0..127
        int c4 = (idx & 15) << 2;         // float col 0..60
        int w  = r >> 4;
        int lr = r & 15;
        float* dst = &sH[w][lr * WS + c4];
        int grow = base + r;
        if (grow < N_NODES) {
            float4 xv = *(const float4*)(x    + (long long)grow * DIM + c4);
            float4 av = *(const float4*)(aggr + (long long)grow * DIM + c4);
            dst[0] = xv.x + av.x;
            dst[1] = xv.y + av.y;
            dst[2] = xv.z + av.z;
            dst[3] = xv.w + av.w;
        } else {
            dst[0] = 0.f; dst[1] = 0.f; dst[2] = 0.f; dst[3] = 0.f;
        }
    }

    const float* const Wg[3] = { W1, W2, W3 };
    const float* const Bg[3] = { b1, b2, b3 };
    float* hin = sH[wid];

#pragma unroll
    for (int layer = 0; layer < 3; ++layer) {
        // stage this layer's weights + bias
        __syncthreads();
        for (int idx = tid; idx < DIM * DIM; idx += 256) {
            sW[(idx >> 6) * WS + (idx & 63)] = Wg[layer][idx];
        }
        if (tid < DIM) sB[tid] = Bg[layer][tid];
        __syncthreads();

        // C init: broadcast bias per output column
        float bv0 = sB[ 0 + l15];
        float bv1 = sB[16 + l15];
        float bv2 = sB[32 + l15];
        float bv3 = sB[48 + l15];
        v8f acc0, acc1, acc2, acc3;
#pragma unroll
        for (int r = 0; r < 8; ++r) {
            acc0[r] = bv0; acc1[r] = bv1; acc2[r] = bv2; acc3[r] = bv3;
        }

        // K loop: 16 steps of K=4; A frag shared across the 4 N-tiles
#pragma unroll
        for (int k0 = 0; k0 < DIM; k0 += 4) {
            int koff = k0 + (lhalf << 1);     // lanes 0-15: K=k0,k0+1 ; 16-31: K=k0+2,k0+3
            v2f a;
            a.x = hin[l15 * WS + koff];
            a.y = hin[l15 * WS + koff + 1];

            v2f fb0, fb1, fb2, fb3;
            fb0.x = sW[( 0 + l15) * WS + koff];  fb0.y = sW[( 0 + l15) * WS + koff + 1];
            fb1.x = sW[(16 + l15) * WS + koff];  fb1.y = sW[(16 + l15) * WS + koff + 1];
            fb2.x = sW[(32 + l15) * WS + koff];  fb2.y = sW[(32 + l15) * WS + koff + 1];
            fb3.x = sW[(48 + l15) * WS + koff];  fb3.y = sW[(48 + l15) * WS + koff + 1];

            acc0 = __builtin_amdgcn_wmma_f32_16x16x4_f32(false, a, false, fb0, (short)0, acc0, false, false);
            acc1 = __builtin_amdgcn_wmma_f32_16x16x4_f32(false, a, false, fb1, (short)0, acc1, false, false);
            acc2 = __builtin_amdgcn_wmma_f32_16x16x4_f32(false, a, false, fb2, (short)0, acc2, false, false);
            acc3 = __builtin_amdgcn_wmma_f32_16x16x4_f32(false, a, false, fb3, (short)0, acc3, false, false);
        }

        if (layer < 2) {
            // ReLU + write back into the same tile (all reads already issued)
#pragma unroll
            for (int r = 0; r < 8; ++r) {
                int row = r + (lhalf << 3);   // C/D layout: M=r (low half) / M=8+r (high half)
                hin[row * WS +  0 + l15] = fmaxf(acc0[r], 0.f);
                hin[row * WS + 16 + l15] = fmaxf(acc1[r], 0.f);
                hin[row * WS + 32 + l15] = fmaxf(acc2[r], 0.f);
                hin[row * WS + 48 + l15] = fmaxf(acc3[r], 0.f);
            }
        } else {
            // final layer: write straight to global output
#pragma unroll
            for (int r = 0; r < 8; ++r) {
                int row  = r + (lhalf << 3);
                int grow = base + wid * 16 + row;
                if (grow < N_NODES) {
                    float* op = out + (long long)grow * DIM;
                    op[ 0 + l15] = acc0[r];
                    op[16 + l15] = acc1[r];
                    op[32 + l15] = acc2[r];
                    op[48 + l15] = acc3[r];
                }
            }
        }
    }
}

extern "C" void kernel_launch(void* const* d_in, const int* in_sizes, int n_in,
                              void* d_out, int out_size, void* d_ws, size_t ws_size,
                              hipStream_t stream) {
    const float*     x   = (const float*)d_in[0];
    const long long* ei  = (const long long*)d_in[1];   // int64 edge_index (2, E)
    const float*     ea  = (const float*)d_in[2];
    const float*     W1  = (const float*)d_in[3];
    const float*     b1  = (const float*)d_in[4];
    const float*     W2  = (const float*)d_in[5];
    const float*     b2  = (const float*)d_in[6];
    const float*     W3  = (const float*)d_in[7];
    const float*     b3  = (const float*)d_in[8];
    float* out  = (float*)d_out;
    float* aggr = (float*)d_ws;                          // N_NODES*DIM floats = 12.8 MB

    // 1) zero aggr
    {
        int n4 = N_NODES * DIM / 4;
        int blocks = (n4 + 255) / 256;
        gine_zero_kernel<<<blocks, 256, 0, stream>>>(aggr, n4);
    }
    // 2) edge messages + scatter add
    {
        long long threads = (long long)N_EDGES * 16;
        int blocks = (int)((threads + 255) / 256);
        gine_edge_kernel<<<blocks, 256, 0, stream>>>(x, ei, ea, aggr);
    }
    // 3) fused residual + 3-layer MLP (WMMA fp32)
    {
        int blocks = (N_NODES + 127) / 128;
        gine_mlp_kernel<<<blocks, 256, 0, stream>>>(x, aggr, W1, b1, W2, b2, W3, b3, out);
    }
}